// DiscreteAutoregressiveFlow_68805376082450
// MI455X (gfx1250) — compile-verified
//
#include <hip/hip_runtime.h>
#include <hip/hip_bf16.h>

// ---------------------------------------------------------------------------
// DiscreteAutoregressiveFlow forward, collapsed algebraically:
//   inputs are exact one-hot [B,S,64]  -> MLP logits depend only on token id
//   -> 64x128 logit table = relu(W1+b1) @ W2 + b2   (f32 WMMA, exact in f32)
//   straight-through one_hot_argmax forward == hard one-hot
//   one_hot_multiply(one-hot, one-hot) -> index (s*t) mod 64
//   one_hot_add (FFT circ-conv) of one-hots -> index (+) mod 64
//   out[b,s,:] = one_hot( (scale_idx[tok]*tok + loc_idx[tok]) mod 64 )
// Dominant cost: streaming 8MB in + 8MB out  (~0.7us @ 23.3 TB/s).
// ---------------------------------------------------------------------------

typedef __attribute__((ext_vector_type(2))) float v2f;
typedef __attribute__((ext_vector_type(8))) float v8f;

#define VOCAB 64
#define HID   4096
#define OUTW  128   // 2*VOCAB

// ---------------------------------------------------------------------------
// Kernel 1: logits[64 x 128] = relu(W1 + b1) @ W2  via V_WMMA_F32_16X16X4_F32.
// Grid (8,4): blockIdx.x = N-tile (8 tiles of 16 over 128 cols),
//             blockIdx.y = M-tile (4 tiles of 16 over 64 token rows).
// One wave (32 lanes) per 16x16 output tile, K=4096 in steps of 4.
// A 16x4 f32 fragment (ISA 7.12.2): lanes 0-15 -> M=lane, v.x=K+0, v.y=K+1;
//                                   lanes 16-31 -> M=lane-16, v.x=K+2, v.y=K+3.
// B 4x16 fragment uses the same K split, N = lane&15 striped across lanes.
// D 16x16 f32: VGPR r -> (M=r, N=lane) lanes 0-15, (M=r+8, N=lane-16) 16-31.
// 4 independent accumulators break the WMMA->WMMA RAW dependency chain.
// ---------------------------------------------------------------------------
__global__ __launch_bounds__(32) void k_table_gemm(
    const float* __restrict__ W1, const float* __restrict__ b1,
    const float* __restrict__ W2, float* __restrict__ logits)
{
  const int lane = threadIdx.x & 31;
  const int tm   = blockIdx.y;
  const int tn   = blockIdx.x;
  const int mrow = lane & 15;
  const int hi   = lane >> 4;                 // 0: K+{0,1}  1: K+{2,3}
  const float* __restrict__ w1row = W1 + (size_t)(tm * 16 + mrow) * HID;
  const int n = tn * 16 + mrow;

  v8f acc[4];
#pragma unroll
  for (int u = 0; u < 4; ++u) acc[u] = (v8f){};

  for (int k = 0; k < HID; k += 16) {
#pragma unroll
    for (int u = 0; u < 4; ++u) {
      const int ka = k + u * 4 + 2 * hi;
      // A fragment: relu(W1[t, ka..ka+1] + b1[ka..ka+1])
      v2f w  = *(const v2f*)(w1row + ka);
      v2f bb = *(const v2f*)(b1 + ka);
      v2f a;
      a.x = fmaxf(w.x + bb.x, 0.0f);
      a.y = fmaxf(w.y + bb.y, 0.0f);
      // B fragment: W2[ka..ka+1, n]  (row stride 128)
      v2f b;
      b.x = W2[(size_t)ka * OUTW + n];
      b.y = W2[((size_t)ka + 1) * OUTW + n];
      acc[u] = __builtin_amdgcn_wmma_f32_16x16x4_f32(
          /*neg_a=*/false, a, /*neg_b=*/false, b,
          /*c_mod=*/(short)0, acc[u], /*reuse_a=*/false, /*reuse_b=*/false);
    }
  }

  v8f c = acc[0] + acc[1] + acc[2] + acc[3];
  const int mbase = tm * 16 + hi * 8;
  const int ncol  = tn * 16 + mrow;
#pragma unroll
  for (int r = 0; r < 8; ++r)
    logits[(size_t)(mbase + r) * OUTW + ncol] = c[r];
}

// ---------------------------------------------------------------------------
// Kernel 2: per token t: loc_idx = argmax(logits[t, 0:64]  + b2[0:64])
//                        scale_idx = argmax(logits[t,64:128] + b2[64:128])
// (first-max tie-break, matching jnp.argmax)
// fidx[t] = (scale_idx * t + loc_idx) mod 64
// ---------------------------------------------------------------------------
__global__ __launch_bounds__(64) void k_argmax(
    const float* __restrict__ logits, const float* __restrict__ b2,
    int* __restrict__ fidx)
{
  const int t = threadIdx.x;
  if (t >= VOCAB) return;
  const float* __restrict__ row = logits + (size_t)t * OUTW;
  float bl = -3.402823466e38f; int li = 0;
  float bs = -3.402823466e38f; int si = 0;
  for (int j = 0; j < VOCAB; ++j) {
    float vl = row[j] + b2[j];
    if (vl > bl) { bl = vl; li = j; }
    float vs = row[VOCAB + j] + b2[VOCAB + j];
    if (vs > bs) { bs = vs; si = j; }
  }
  fidx[t] = (si * t + li) & (VOCAB - 1);
}

// ---------------------------------------------------------------------------
// Kernel 3: bandwidth-bound emit. One wave32 per [B*S] row.
// Recover token from the one-hot row (2 NT loads/lane), OR-reduce across the
// wave with shfl_xor (exactly one lane contributes a nonzero token), then
// write the output one-hot with non-temporal stores (streamed, no reuse).
// ---------------------------------------------------------------------------
__global__ __launch_bounds__(256) void k_emit(
    const float* __restrict__ inputs, const int* __restrict__ fidx,
    float* __restrict__ out, int nrows)
{
  const int gwave = (int)((blockIdx.x * 256u + threadIdx.x) >> 5);
  const int lane  = threadIdx.x & 31;
  if (gwave >= nrows) return;

  const float* __restrict__ in = inputs + (size_t)gwave * VOCAB;
  const float v0 = __builtin_nontemporal_load(in + lane);
  const float v1 = __builtin_nontemporal_load(in + lane + 32);
  int tok = (v0 > 0.5f) ? lane : ((v1 > 0.5f) ? (lane + 32) : 0);
#pragma unroll
  for (int off = 16; off > 0; off >>= 1)
    tok |= __shfl_xor(tok, off, 32);          // wave32 OR-reduce

  const int f = fidx[tok];
  float* __restrict__ o = out + (size_t)gwave * VOCAB;
  __builtin_nontemporal_store((lane == f)      ? 1.0f : 0.0f, o + lane);
  __builtin_nontemporal_store((lane + 32 == f) ? 1.0f : 0.0f, o + lane + 32);
}

// ---------------------------------------------------------------------------
extern "C" void kernel_launch(void* const* d_in, const int* in_sizes, int n_in,
                              void* d_out, int out_size, void* d_ws, size_t ws_size,
                              hipStream_t stream) {
  const float* inputs = (const float*)d_in[0];   // [B,S,64] one-hot f32
  const float* W1     = (const float*)d_in[1];   // [64,4096]
  const float* b1     = (const float*)d_in[2];   // [4096]
  const float* W2     = (const float*)d_in[3];   // [4096,128]
  const float* b2     = (const float*)d_in[4];   // [128]
  float*       out    = (float*)d_out;           // [B,S,64]

  float* logits = (float*)d_ws;                                   // 64*128 f32
  int*   fidx   = (int*)((char*)d_ws + VOCAB * OUTW * sizeof(float)); // 64 i32

  const int nrows = in_sizes[0] / VOCAB;         // B*S

  k_table_gemm<<<dim3(8, 4, 1), 32, 0, stream>>>(W1, b1, W2, logits);
  k_argmax<<<1, 64, 0, stream>>>(logits, b2, fidx);

  const int blocks = (nrows * 32 + 255) / 256;   // one wave32 per row
  k_emit<<<blocks, 256, 0, stream>>>(inputs, fidx, out, nrows);
}